// AdjacencyErrorAwareLoss_24378234372149
// MI455X (gfx1250) — compile-verified
//
#include <hip/hip_runtime.h>

typedef __attribute__((ext_vector_type(2))) float v2f;
typedef __attribute__((ext_vector_type(8))) float v8f;

#define Q 128
#define B 64
#define LDSW 132   // padded LDS row stride (floats): 132*4 = 528 = 33*16 -> float4-aligned rows, conflict-free

// ---------------- Kernel 1: A_fid = (d_hw == 1) * max(1 - d_error, 0) ----------------
__global__ void k_build_afid(const float* __restrict__ d_hw,
                             const float* __restrict__ d_err,
                             float* __restrict__ afid) {
    int idx = blockIdx.x * blockDim.x + threadIdx.x;
    if (idx < Q * Q) {
        afid[idx] = (d_hw[idx] == 1.0f) ? fmaxf(1.0f - d_err[idx], 0.0f) : 0.0f;
    }
}

// ---------------- Kernel 2: Cmean = (1/B) * sum_b  P_b * A * P_b^T  (f32 WMMA) ----------------
// Grid: 16 blocks (4x4 tiling of the 128x128 output, 32x32 tile per block), 128 threads (4 waves).
__global__ __launch_bounds__(128)
void k_bilinear_wmma(const float* __restrict__ P,     // [B][Q][Q]
                     const float* __restrict__ A,     // [Q][Q]  (A_fid, L2-resident)
                     float* __restrict__ cmean) {     // [Q][Q]
    __shared__ __align__(16) float Pi_s[32 * LDSW];   // P rows bi*32 .. bi*32+31
    __shared__ __align__(16) float Pj_s[32 * LDSW];   // P rows bj*32 .. bj*32+31
    __shared__ __align__(16) float Ts  [32 * LDSW];   // T = Pi * A  (32 x 128 strip)

    const int bi   = blockIdx.x >> 2;       // 0..3  -> output rows  bi*32..
    const int bj   = blockIdx.x & 3;        // 0..3  -> output cols  bj*32..
    const int tid  = threadIdx.x;
    const int wave = tid >> 5;              // 0..3
    const int lane = tid & 31;
    const int m    = lane & 15;             // A-matrix row / B-matrix col / D col
    const int kb   = (lane >> 4) << 1;      // K sub-offset: lanes 0-15 -> 0, lanes 16-31 -> 2
    const int rb   = (lane >> 4) << 3;      // D-row sub-offset: 0 or 8

    // Second-GEMM accumulator: one 16x16 f32 tile per wave, held across all batches.
    const int mr = wave >> 1;               // 0..1
    const int mc = wave & 1;                // 0..1
    v8f accM = {};

    for (int b = 0; b < B; ++b) {
        const float* Pb = P + b * (Q * Q);

        // ---- load P strips into LDS (float4, padded rows) ----
        for (int t = tid; t < 32 * 32; t += 128) {        // 32 rows x 32 float4
            int row = t >> 5, c4 = t & 31;
            float4 vi = ((const float4*)(Pb + (bi * 32 + row) * Q))[c4];
            float4 vj = ((const float4*)(Pb + (bj * 32 + row) * Q))[c4];
            *(float4*)&Pi_s[row * LDSW + c4 * 4] = vi;
            *(float4*)&Pj_s[row * LDSW + c4 * 4] = vj;
        }
        __syncthreads();

        // ---- Phase 1: T = Pi * A  (32x128 strip = 2x8 tiles of 16x16; 4 tiles per wave) ----
        for (int t = wave * 4; t < wave * 4 + 4; ++t) {
            const int tr = t >> 3;          // 0..1
            const int tc = t & 7;           // 0..7
            v8f acc = {};
            for (int k0 = 0; k0 < Q; k0 += 4) {
                // A-operand: Pi[tr*16+m, k0+kb], k consecutive -> ds_read_b64
                v2f av = *(const v2f*)&Pi_s[(tr * 16 + m) * LDSW + k0 + kb];
                // B-operand: A[k0+kb, tc*16+m] (global, L2-hot)
                v2f bv;
                bv.x = A[(k0 + kb    ) * Q + tc * 16 + m];
                bv.y = A[(k0 + kb + 1) * Q + tc * 16 + m];
                acc = __builtin_amdgcn_wmma_f32_16x16x4_f32(
                          false, av, false, bv, (short)0, acc, false, false);
            }
            #pragma unroll
            for (int r = 0; r < 8; ++r)     // D layout: VGPR r -> row r (lanes 0-15) / r+8 (16-31)
                Ts[(tr * 16 + rb + r) * LDSW + tc * 16 + m] = acc[r];
        }
        __syncthreads();

        // ---- Phase 2: M-tile += T * Pj^T  (B[k,n] = Pj[n,k] -> consecutive k, ds_read_b64) ----
        for (int k0 = 0; k0 < Q; k0 += 4) {
            v2f av = *(const v2f*)&Ts  [(mr * 16 + m) * LDSW + k0 + kb];
            v2f bv = *(const v2f*)&Pj_s[(mc * 16 + m) * LDSW + k0 + kb];
            accM = __builtin_amdgcn_wmma_f32_16x16x4_f32(
                       false, av, false, bv, (short)0, accM, false, false);
        }
        __syncthreads();
    }

    // ---- write Cmean tile ----
    #pragma unroll
    for (int r = 0; r < 8; ++r) {
        int row = bi * 32 + mr * 16 + rb + r;
        int col = bj * 32 + mc * 16 + m;
        cmean[row * Q + col] = accM[r] * (1.0f / (float)B);
    }
}

// ---------------- Kernel 3: per-block partial sums of w_e * Cmean[i,j] and w_e ----------------
__global__ void k_edge_partials(const float* __restrict__ cmean,
                                const int* __restrict__ pairs,   // [E][2]
                                const float* __restrict__ w,     // [E]
                                int nE,
                                float* __restrict__ nump,
                                float* __restrict__ wsump) {
    __shared__ float s1[256];
    __shared__ float s2[256];
    int tid = threadIdx.x;
    int e = blockIdx.x * 256 + tid;
    float v = 0.0f, we = 0.0f;
    if (e < nE) {
        int i = pairs[2 * e];
        int j = pairs[2 * e + 1];
        we = w[e];
        v = we * cmean[i * Q + j];
    }
    s1[tid] = v;
    s2[tid] = we;
    __syncthreads();
    for (int s = 128; s > 0; s >>= 1) {
        if (tid < s) { s1[tid] += s1[tid + s]; s2[tid] += s2[tid + s]; }
        __syncthreads();
    }
    if (tid == 0) { nump[blockIdx.x] = s1[0]; wsump[blockIdx.x] = s2[0]; }
}

// ---------------- Kernel 4: final reduction, loss = -num / max(wsum, 1e-8) ----------------
__global__ void k_finalize(const float* __restrict__ nump,
                           const float* __restrict__ wsump,
                           int nblk,
                           float* __restrict__ out) {
    __shared__ float s1[64];
    __shared__ float s2[64];
    int t = threadIdx.x;
    s1[t] = (t < nblk) ? nump[t] : 0.0f;
    s2[t] = (t < nblk) ? wsump[t] : 0.0f;
    __syncthreads();
    for (int s = 32; s > 0; s >>= 1) {
        if (t < s) { s1[t] += s1[t + s]; s2[t] += s2[t + s]; }
        __syncthreads();
    }
    if (t == 0) out[0] = -s1[0] / fmaxf(s2[0], 1e-8f);
}

extern "C" void kernel_launch(void* const* d_in, const int* in_sizes, int n_in,
                              void* d_out, int out_size, void* d_ws, size_t ws_size,
                              hipStream_t stream) {
    const float* P     = (const float*)d_in[0];   // [64][128][128] f32
    const float* d_hw  = (const float*)d_in[1];   // [128][128] f32
    const float* d_err = (const float*)d_in[2];   // [128][128] f32
    const int*   pairs = (const int*)  d_in[3];   // [E][2] i32
    const float* w     = (const float*)d_in[4];   // [E] f32
    const int nE = in_sizes[4];                   // 16384
    const int nblk = (nE + 255) / 256;            // 64

    float* ws    = (float*)d_ws;
    float* afid  = ws;                 // 16384 floats
    float* cmean = ws + 16384;         // 16384 floats
    float* nump  = ws + 32768;         // nblk floats
    float* wsump = ws + 32768 + 64;    // nblk floats

    k_build_afid   <<<(Q * Q + 255) / 256, 256, 0, stream>>>(d_hw, d_err, afid);
    k_bilinear_wmma<<<16, 128, 0, stream>>>(P, afid, cmean);
    k_edge_partials<<<nblk, 256, 0, stream>>>(cmean, pairs, w, nE, nump, wsump);
    k_finalize     <<<1, 64, 0, stream>>>(nump, wsump, nblk, (float*)d_out);
}